// SRSChannelEstimator_83184926589386
// MI455X (gfx1250) — compile-verified
//
#include <hip/hip_runtime.h>
#include <math.h>

#define UPN 8
#define BN 32
#define AN 64
#define LN 1632
#define KN 12
#define REDN 408      // LN / LOCC
#define BLKN 12
#define NBLK 136      // LN / BLKN
#define SEARCHN 10
#define NWIN 21
#define RHOF 0.9f

typedef float v2f __attribute__((ext_vector_type(2)));
typedef float v8f __attribute__((ext_vector_type(8)));

// D = A(16x4 f32) * B(4x16 f32) + C(16x16 f32), full fp32 — matches reference precision.
static __device__ inline v8f wmma_f32_16x16x4(v2f a, v2f b, v8f c) {
  return __builtin_amdgcn_wmma_f32_16x16x4_f32(false, a, false, b, (short)0, c, false, false);
}

// ---------------------------------------------------------------------------
// Kernel 1: per (u,b,a): 21-bin DFT (replaces full IFFT; only the window is
// needed for argmax) + noise partial sum. Atomic-accumulate into ws.
// ---------------------------------------------------------------------------
__global__ __launch_bounds__(256)
void k_peak_noise(const float* __restrict__ lsr, const float* __restrict__ lsi,
                  const int* __restrict__ shifts,
                  float* __restrict__ pwr, float* __restrict__ noise) {
  const int blk = blockIdx.x;            // ((u*BN + b) * AN + a)
  const int ub  = blk >> 6;
  const int u   = ub >> 5;
  const int tid = threadIdx.x;
  const float C2PI = 6.28318530717958647692f / (float)LN;
  const int sh    = shifts[u];
  const int ideal = ((KN - sh) % KN) * (LN / KN);

  const float* pr = lsr + (size_t)blk * LN;
  const float* pi = lsi + (size_t)blk * LN;

  float accr[NWIN], acci[NWIN], rotr[NWIN], roti[NWIN], str[NWIN], sti[NWIN];
#pragma unroll
  for (int j = 0; j < NWIN; ++j) {
    int t = ideal + (j - SEARCHN);
    t = ((t % LN) + LN) % LN;
    const float a0 = C2PI * (float)((tid * t) % LN);
    const float as = C2PI * (float)((256 * t) % LN);
    __sincosf(a0, &roti[j], &rotr[j]);   // e^{2pi i * tid * t / L}
    __sincosf(as, &sti[j], &str[j]);     // per-step rotation e^{2pi i * 256 t / L}
    accr[j] = 0.f; acci[j] = 0.f;
  }
  for (int n = tid; n < LN; n += 256) {
    const float xr = pr[n], xi = pi[n];
#pragma unroll
    for (int j = 0; j < NWIN; ++j) {
      accr[j] += xr * rotr[j] - xi * roti[j];
      acci[j] += xr * roti[j] + xi * rotr[j];
      const float nr = rotr[j] * str[j] - roti[j] * sti[j];
      const float ni = rotr[j] * sti[j] + roti[j] * str[j];
      rotr[j] = nr; roti[j] = ni;
    }
  }
  // noise partial: sum |ls[n+1]-ls[n]|^2
  float nacc = 0.f;
  for (int n = tid; n < LN - 1; n += 256) {
    const float dr = pr[n + 1] - pr[n];
    const float di = pi[n + 1] - pi[n];
    nacc += dr * dr + di * di;
  }
  // wave32 shuffle reduction
#pragma unroll
  for (int off = 16; off > 0; off >>= 1) {
#pragma unroll
    for (int j = 0; j < NWIN; ++j) {
      accr[j] += __shfl_down(accr[j], off, 32);
      acci[j] += __shfl_down(acci[j], off, 32);
    }
    nacc += __shfl_down(nacc, off, 32);
  }
  __shared__ float sred[8][2 * NWIN + 1];
  const int wv = tid >> 5, ln = tid & 31;
  if (ln == 0) {
#pragma unroll
    for (int j = 0; j < NWIN; ++j) { sred[wv][j] = accr[j]; sred[wv][NWIN + j] = acci[j]; }
    sred[wv][2 * NWIN] = nacc;
  }
  __syncthreads();
  if (tid < NWIN) {
    float sr = 0.f, si = 0.f;
#pragma unroll
    for (int w = 0; w < 8; ++w) { sr += sred[w][tid]; si += sred[w][NWIN + tid]; }
    atomicAdd(&pwr[ub * NWIN + tid], sr * sr + si * si);   // |H|^2 summed over antennas
  } else if (tid == NWIN) {
    float sn = 0.f;
#pragma unroll
    for (int w = 0; w < 8; ++w) sn += sred[w][2 * NWIN];
    atomicAdd(&noise[ub], sn);
  }
}

// ---------------------------------------------------------------------------
// Kernel 2: per (u,b): argmax over 21 window bins (first max wins, matching
// jnp.argmax), compute m = toff + ideal_peak; finalize noise power.
// ---------------------------------------------------------------------------
__global__ __launch_bounds__(256)
void k_select(const float* __restrict__ pwr, const float* __restrict__ noise,
              const int* __restrict__ shifts,
              int* __restrict__ m_val, float* __restrict__ noise_pw) {
  const int t = threadIdx.x;               // ub = u*BN + b
  if (t >= UPN * BN) return;
  const int u = t >> 5;
  const int sh = shifts[u];
  const int ideal = ((KN - sh) % KN) * (LN / KN);
  float best = pwr[t * NWIN];
  int bj = 0;
  for (int j = 1; j < NWIN; ++j) {
    const float v = pwr[t * NWIN + j];
    if (v > best) { best = v; bj = j; }    // strict > keeps first occurrence
  }
  m_val[t] = ideal + (bj - SEARCHN);
  noise_pw[t] = noise[t] / ((float)AN * (float)(LN - 1)) * 0.5f;
}

// ---------------------------------------------------------------------------
// Kernel 3: per (u,b): W = C * inv(C + s*I), 12x12, Gauss-Jordan in LDS
// (C + sI is SPD -> no pivoting needed).
// ---------------------------------------------------------------------------
__global__ __launch_bounds__(64)
void k_wfilter(const float* __restrict__ noise_pw, float* __restrict__ Wbuf) {
  __shared__ float M[64][145];    // padded row stride to dodge bank conflicts
  __shared__ float srp[BLKN];
  const int lt = threadIdx.x;
  const int g  = blockIdx.x * 64 + lt;
  if (lt == 0) {
    srp[0] = 1.f;
    for (int d = 1; d < BLKN; ++d) srp[d] = srp[d - 1] * RHOF;
  }
  __syncthreads();
  const float s = noise_pw[g];
  float* Mm = M[lt];
  for (int i = 0; i < BLKN; ++i)
    for (int j = 0; j < BLKN; ++j)
      Mm[i * BLKN + j] = srp[i > j ? i - j : j - i] + (i == j ? s : 0.f);
  // in-place Gauss-Jordan inversion
  for (int k = 0; k < BLKN; ++k) {
    const float ip = 1.f / Mm[k * BLKN + k];
    Mm[k * BLKN + k] = 1.f;
    for (int j = 0; j < BLKN; ++j) Mm[k * BLKN + j] *= ip;
    for (int i = 0; i < BLKN; ++i) {
      if (i == k) continue;
      const float f = Mm[i * BLKN + k];
      Mm[i * BLKN + k] = 0.f;
      for (int j = 0; j < BLKN; ++j) Mm[i * BLKN + j] -= f * Mm[k * BLKN + j];
    }
  }
  // W = C * inv
  for (int i = 0; i < BLKN; ++i)
    for (int j = 0; j < BLKN; ++j) {
      float acc = 0.f;
      for (int k = 0; k < BLKN; ++k)
        acc += srp[i > k ? i - k : k - i] * Mm[k * BLKN + j];
      Wbuf[(size_t)g * 144 + i * BLKN + j] = acc;
    }
}

// ---------------------------------------------------------------------------
// Kernel 4: per (u,b, group of 8 antennas):
//   phase-rotate ls by e^{2pi i m n / L}, average groups of 4 -> h_avg (LDS),
//   then per WMMA tile build A from on-the-fly linear interpolation and apply
//   the 12x12 W filter via V_WMMA_F32_16X16X4_F32 (K=12 as 3 chunks of 4,
//   N padded 12->16). Writes interleaved (re,im) output.
// ---------------------------------------------------------------------------
__global__ __launch_bounds__(256)
void k_main(const float* __restrict__ lsr, const float* __restrict__ lsi,
            const int* __restrict__ m_val, const float* __restrict__ Wbuf,
            float* __restrict__ out) {
  __shared__ float s_re[8][REDN];
  __shared__ float s_im[8][REDN];
  __shared__ float s_W[16][BLKN];     // rows 12..15 zero-padded
  const int blk = blockIdx.x;         // ub*8 + antenna-group
  const int grp = blk & 7;
  const int ub  = blk >> 3;
  const int a0  = grp * 8;
  const int tid = threadIdx.x;
  const float C2PI = 6.28318530717958647692f / (float)LN;

  if (tid < 16 * BLKN) {
    const int i = tid / BLKN, j = tid - i * BLKN;
    s_W[i][j] = (i < BLKN) ? Wbuf[(size_t)ub * 144 + i * BLKN + j] : 0.f;
  }

  const int m  = m_val[ub];
  const int mm = ((m % LN) + LN) % LN;

  const float* br = lsr + ((size_t)ub * AN + a0) * LN;
  const float* bi = lsi + ((size_t)ub * AN + a0) * LN;

  // Phase A: h_avg = mean of 4 phase-rotated samples
  for (int al = 0; al < 8; ++al) {
    const float* pr = br + al * LN;
    const float* pi = bi + al * LN;
    __builtin_prefetch(pr + LN, 0, 1);   // global_prefetch_b8 for next antenna row
    __builtin_prefetch(pi + LN, 0, 1);
    for (int r = tid; r < REDN; r += 256) {
      const float4 vr = *(const float4*)(pr + 4 * r);
      const float4 vi = *(const float4*)(pi + 4 * r);
      float ar = 0.f, ai = 0.f;
#pragma unroll
      for (int q = 0; q < 4; ++q) {
        const int n = 4 * r + q;
        float sn, cs;
        __sincosf(C2PI * (float)((mm * n) % LN), &sn, &cs);
        const float xr = (q == 0) ? vr.x : (q == 1) ? vr.y : (q == 2) ? vr.z : vr.w;
        const float xi = (q == 0) ? vi.x : (q == 1) ? vi.y : (q == 2) ? vi.z : vi.w;
        ar += xr * cs - xi * sn;
        ai += xr * sn + xi * cs;
      }
      s_re[al][r] = ar * 0.25f;
      s_im[al][r] = ai * 0.25f;
    }
  }
  __syncthreads();

  const int wv   = tid >> 5;
  const int l    = tid & 31;
  const int half = (l >> 4) & 1;
  const int li   = l & 15;

  // B fragment: B[k][i] = W[i][k0+k].  Layout: VGPR0 holds K=k0+2*half,
  // VGPR1 holds K=k0+2*half+1, column N = lane%16.
  v2f bfrag[3];
#pragma unroll
  for (int c = 0; c < 3; ++c) {
    const int k0 = 4 * c + 2 * half;
    bfrag[c].x = s_W[li][k0];
    bfrag[c].y = s_W[li][k0 + 1];
  }

  // 8 antennas * 136 blocks = 1088 rows = 68 tiles of 16; one tile per wave pass.
  for (int t = wv; t < 68; t += 8) {
    const int row = t * 16 + li;       // A row M = lane%16 for both lane halves
    const int al  = row / NBLK;
    const int nb  = (row - al * NBLK) * BLKN;
    v2f are[3], aim[3];
#pragma unroll
    for (int c = 0; c < 3; ++c) {
      const int k0 = 4 * c + 2 * half;
#pragma unroll
      for (int q = 0; q < 2; ++q) {
        const int n = nb + k0 + q;
        float p = ((float)n - 1.5f) * 0.25f;          // (t - (LOCC-1)/2) / LOCC
        p = fminf(fmaxf(p, 0.f), 407.f);
        int i0 = (int)p;                              // floor (p >= 0)
        if (i0 > 406) i0 = 406;
        const float fr = p - (float)i0;
        const float re = s_re[al][i0] + (s_re[al][i0 + 1] - s_re[al][i0]) * fr;
        const float im = s_im[al][i0] + (s_im[al][i0 + 1] - s_im[al][i0]) * fr;
        if (q == 0) { are[c].x = re; aim[c].x = im; }
        else        { are[c].y = re; aim[c].y = im; }
      }
    }
    v8f cre = {0.f, 0.f, 0.f, 0.f, 0.f, 0.f, 0.f, 0.f};
    v8f cim = {0.f, 0.f, 0.f, 0.f, 0.f, 0.f, 0.f, 0.f};
#pragma unroll
    for (int c = 0; c < 3; ++c) {
      cre = wmma_f32_16x16x4(are[c], bfrag[c], cre);
      cim = wmma_f32_16x16x4(aim[c], bfrag[c], cim);
    }
    if (li < BLKN) {                                   // skip padded columns 12..15
#pragma unroll
      for (int v = 0; v < 8; ++v) {
        const int M   = v + 8 * half;                  // C/D layout: VGPR v -> M = v + 8*(lane>=16)
        const int r2  = t * 16 + M;
        const int al2 = r2 / NBLK;
        const int n   = (r2 - al2 * NBLK) * BLKN + li;
        float2* o = (float2*)out + (((size_t)ub * AN + a0 + al2) * LN + n);
        *o = make_float2(cre[v], cim[v]);
      }
    }
  }
}

// ---------------------------------------------------------------------------
extern "C" void kernel_launch(void* const* d_in, const int* in_sizes, int n_in,
                              void* d_out, int out_size, void* d_ws, size_t ws_size,
                              hipStream_t stream) {
  (void)in_sizes; (void)n_in; (void)out_size; (void)ws_size;
  const float* lsr   = (const float*)d_in[0];   // [8,32,64,1632]
  const float* lsi   = (const float*)d_in[1];   // [8,32,64,1632]
  const int* shifts  = (const int*)d_in[2];     // [8]
  float* out = (float*)d_out;                   // [8,32,64,1632,2]

  float* wsf      = (float*)d_ws;
  float* pwr      = wsf;            // 256*21 floats
  float* noise    = wsf + 5376;     // 256
  float* noise_pw = wsf + 5632;     // 256
  int*   m_val    = (int*)(wsf + 5888);  // 256
  float* Wbuf     = wsf + 6144;     // 256*144

  hipMemsetAsync(d_ws, 0, 5632 * sizeof(float), stream);  // zero pwr + noise accumulators
  k_peak_noise<<<UPN * BN * AN, 256, 0, stream>>>(lsr, lsi, shifts, pwr, noise);
  k_select<<<1, 256, 0, stream>>>(pwr, noise, shifts, m_val, noise_pw);
  k_wfilter<<<4, 64, 0, stream>>>(noise_pw, Wbuf);
  k_main<<<UPN * BN * 8, 256, 0, stream>>>(lsr, lsi, m_val, Wbuf, out);
}